// ScEncoder_78125455114509
// MI455X (gfx1250) — compile-verified
//
#include <hip/hip_runtime.h>
#include <cstdint>

#define NNODES 50000
#define NEDGES 300000
#define NLAYER 2
#define NREL   2
#define NGRAPH 4
#define DDIM   128

typedef __attribute__((ext_vector_type(2))) float v2f;
typedef __attribute__((ext_vector_type(8))) float v8f;

__device__ __forceinline__ unsigned mapOrderedF32(float f) {
  unsigned u = __float_as_uint(f);
  return (u & 0x80000000u) ? ~u : (u | 0x80000000u);
}
__device__ __forceinline__ float unmapOrderedF32(unsigned u) {
  return __uint_as_float((u & 0x80000000u) ? (u & 0x7fffffffu) : ~u);
}

// ---------------------------------------------------------------------------
// Generic zero kernel
// ---------------------------------------------------------------------------
__global__ void k_zero(float* __restrict__ p, long n) {
  long i = (long)blockIdx.x * blockDim.x + threadIdx.x;
  if (i < n) p[i] = 0.0f;
}

// ---------------------------------------------------------------------------
// Projection GEMMs: el[g] = x[layer(g)] @ Wsrc[g] + bsrc[g]
//                   er[g] = x[layer(g)] @ Wdst[g] + bdst[g]
// blockIdx.y in [0,8): g = y>>1, which = y&1.  256 threads = 8 waves,
// each wave owns a 16-row tile, chains 32x V_WMMA_F32_16X16X4_F32 per
// 16x16 output tile (8 col tiles), W staged in 64KB LDS.
// ---------------------------------------------------------------------------
__global__ __launch_bounds__(256) void k_gemm_el_er(
    const float* __restrict__ x,
    const float* __restrict__ Wsrc, const float* __restrict__ bsrc,
    const float* __restrict__ Wdst, const float* __restrict__ bdst,
    float* __restrict__ el, float* __restrict__ er) {
  __shared__ float lW[DDIM * DDIM];
  const int mid = blockIdx.y;
  const int g = mid >> 1, which = mid & 1, layer = g >> 1;
  const float* A    = x + (size_t)layer * NNODES * DDIM;
  const float* W    = (which ? Wdst : Wsrc) + (size_t)g * DDIM * DDIM;
  const float* bias = (which ? bdst : bsrc) + (size_t)g * DDIM;
  float* out        = (which ? er : el) + (size_t)g * NNODES * DDIM;

  for (int i = threadIdx.x; i < DDIM * DDIM; i += 256) lW[i] = W[i];
  __syncthreads();

  const int wave = threadIdx.x >> 5, lane = threadIdx.x & 31;
  const int lhalf = lane >> 4, lrow = lane & 15;

  for (int tile0 = blockIdx.x * 128; tile0 < NNODES; tile0 += gridDim.x * 128) {
    const int rowBase = tile0 + wave * 16;
    if (rowBase >= NNODES) continue;
    // A fragments for K = 0..127 (16x4 fp32 tiles): 64 VGPRs
    v2f a[32];
    const float* ar = A + (size_t)(rowBase + lrow) * DDIM + 2 * lhalf;
#pragma unroll
    for (int s = 0; s < 32; ++s) a[s] = *(const v2f*)(ar + 4 * s);

#pragma unroll 1
    for (int c = 0; c < 8; ++c) {
      const float b0 = bias[c * 16 + lrow];
      v8f acc = {b0, b0, b0, b0, b0, b0, b0, b0};
      const float* wp = lW + c * 16 + lrow + (2 * lhalf) * DDIM;
#pragma unroll
      for (int s = 0; s < 32; ++s) {
        v2f b;
        b.x = wp[(4 * s)     * DDIM];
        b.y = wp[(4 * s + 1) * DDIM];
        acc = __builtin_amdgcn_wmma_f32_16x16x4_f32(
            false, a[s], false, b, (short)0, acc, false, false);
      }
      float* op = out + (size_t)(rowBase + lhalf * 8) * DDIM + c * 16 + lrow;
#pragma unroll
      for (int r = 0; r < 8; ++r) op[(size_t)r * DDIM] = acc[r];
    }
  }
}

// ---------------------------------------------------------------------------
// Semantic-attention GEMM + tanh + mean-over-nodes reduction:
//   spAcc[y] += sum_n tanh( A[y] @ W[y/wdiv] + b[y/wdiv] )   (divide by N later)
// ---------------------------------------------------------------------------
__global__ __launch_bounds__(256) void k_gemm_tanh_reduce(
    const float* __restrict__ Abase, const float* __restrict__ Wbase,
    const float* __restrict__ bbase, float* __restrict__ spAcc, int wdiv) {
  __shared__ float lW[DDIM * DDIM];
  const int y = blockIdx.y;
  const float* A    = Abase + (size_t)y * NNODES * DDIM;
  const int wi      = y / wdiv;
  const float* W    = Wbase + (size_t)wi * DDIM * DDIM;
  const float* bias = bbase + (size_t)wi * DDIM;
  float* sp         = spAcc + (size_t)y * DDIM;

  for (int i = threadIdx.x; i < DDIM * DDIM; i += 256) lW[i] = W[i];
  __syncthreads();

  const int wave = threadIdx.x >> 5, lane = threadIdx.x & 31;
  const int lhalf = lane >> 4, lrow = lane & 15;

  for (int tile0 = blockIdx.x * 128; tile0 < NNODES; tile0 += gridDim.x * 128) {
    const int rowBase = tile0 + wave * 16;
    if (rowBase >= NNODES) continue;
    v2f a[32];
    const float* ar = A + (size_t)(rowBase + lrow) * DDIM + 2 * lhalf;
#pragma unroll
    for (int s = 0; s < 32; ++s) a[s] = *(const v2f*)(ar + 4 * s);

#pragma unroll 1
    for (int c = 0; c < 8; ++c) {
      const float b0 = bias[c * 16 + lrow];
      v8f acc = {b0, b0, b0, b0, b0, b0, b0, b0};
      const float* wp = lW + c * 16 + lrow + (2 * lhalf) * DDIM;
#pragma unroll
      for (int s = 0; s < 32; ++s) {
        v2f b;
        b.x = wp[(4 * s)     * DDIM];
        b.y = wp[(4 * s + 1) * DDIM];
        acc = __builtin_amdgcn_wmma_f32_16x16x4_f32(
            false, a[s], false, b, (short)0, acc, false, false);
      }
      float sme = 0.0f;
#pragma unroll
      for (int r = 0; r < 8; ++r) sme += tanhf(acc[r]);
      sme += __shfl_xor(sme, 16);         // combine row halves
      if (lhalf == 0) unsafeAtomicAdd(&sp[c * 16 + lrow], sme);
    }
  }
}

// ---------------------------------------------------------------------------
// Edge attention logits + running segment max (one wave per edge)
// ---------------------------------------------------------------------------
__global__ __launch_bounds__(256) void k_edge_logits(
    const float* __restrict__ el, const float* __restrict__ er,
    const int* __restrict__ src, const int* __restrict__ dst,
    const float* __restrict__ attn, float* __restrict__ elog,
    unsigned* __restrict__ emaxU) {
  const long tid  = (long)blockIdx.x * blockDim.x + threadIdx.x;
  const long w    = tid >> 5;
  const int lane  = threadIdx.x & 31;
  if (w >= (long)NGRAPH * NEDGES) return;
  const int g = (int)(w / NEDGES), e = (int)(w % NEDGES);
  const long eidx = (long)g * NEDGES + e;
  const int s = src[eidx], d = dst[eidx];

  const float4 av = *(const float4*)(el + ((size_t)g * NNODES + s) * DDIM + lane * 4);
  const float4 bv = *(const float4*)(er + ((size_t)g * NNODES + d) * DDIM + lane * 4);
  const float4 wv = *(const float4*)(attn + (size_t)g * DDIM + lane * 4);

  float m, dot = 0.0f;
  m = av.x + bv.x; dot += (m > 0.0f ? m : 0.2f * m) * wv.x;
  m = av.y + bv.y; dot += (m > 0.0f ? m : 0.2f * m) * wv.y;
  m = av.z + bv.z; dot += (m > 0.0f ? m : 0.2f * m) * wv.z;
  m = av.w + bv.w; dot += (m > 0.0f ? m : 0.2f * m) * wv.w;
#pragma unroll
  for (int off = 16; off > 0; off >>= 1) dot += __shfl_xor(dot, off);

  if (lane == 0) {
    elog[eidx] = dot;
    atomicMax(&emaxU[(size_t)g * NNODES + d], mapOrderedF32(dot));
  }
}

// ---------------------------------------------------------------------------
// a = exp(e - emax[dst]); denom[dst] += a   (one thread per edge)
// ---------------------------------------------------------------------------
__global__ void k_edge_exp(float* __restrict__ elog, const int* __restrict__ dst,
                           const unsigned* __restrict__ emaxU,
                           float* __restrict__ denom) {
  const long i = (long)blockIdx.x * blockDim.x + threadIdx.x;
  if (i >= (long)NGRAPH * NEDGES) return;
  const int g = (int)(i / NEDGES);
  const int d = dst[i];
  const float mx = unmapOrderedF32(emaxU[(size_t)g * NNODES + d]);
  const float a = __expf(elog[i] - mx);
  elog[i] = a;
  unsafeAtomicAdd(&denom[(size_t)g * NNODES + d], a);
}

// ---------------------------------------------------------------------------
// h[dst] += (a/denom[dst]) * el[src]   (one wave per edge, f32 atomics)
// ---------------------------------------------------------------------------
__global__ __launch_bounds__(256) void k_edge_aggregate(
    const float* __restrict__ el, const float* __restrict__ elog,
    const int* __restrict__ src, const int* __restrict__ dst,
    const float* __restrict__ denom, float* __restrict__ h) {
  const long tid = (long)blockIdx.x * blockDim.x + threadIdx.x;
  const long w   = tid >> 5;
  const int lane = threadIdx.x & 31;
  if (w >= (long)NGRAPH * NEDGES) return;
  const int g = (int)(w / NEDGES), e = (int)(w % NEDGES);
  const long eidx = (long)g * NEDGES + e;
  const int s = src[eidx], d = dst[eidx];
  const float alpha = elog[eidx] / denom[(size_t)g * NNODES + d];

  const float4 av = *(const float4*)(el + ((size_t)g * NNODES + s) * DDIM + lane * 4);
  float* hp = h + ((size_t)g * NNODES + d) * DDIM + lane * 4;
  unsafeAtomicAdd(hp + 0, alpha * av.x);
  unsafeAtomicAdd(hp + 1, alpha * av.y);
  unsafeAtomicAdd(hp + 2, alpha * av.z);
  unsafeAtomicAdd(hp + 3, alpha * av.w);
}

__global__ void k_elu(float* __restrict__ h, long n) {
  const long i = (long)blockIdx.x * blockDim.x + threadIdx.x;
  if (i >= n) return;
  const float v = h[i];
  h[i] = v > 0.0f ? v : expm1f(v);
}

// ---------------------------------------------------------------------------
// beta = softmax_p( (spAcc[gi*P+p]/N) . att[gi] )  (tiny, one block)
// ---------------------------------------------------------------------------
__global__ void k_beta(const float* __restrict__ spAcc, const float* __restrict__ att,
                       float* __restrict__ betaOut, int P, int ngroups) {
  __shared__ float red[128];
  __shared__ float sc[8];
  const int t = threadIdx.x;
  for (int gi = 0; gi < ngroups; ++gi) {
    for (int p = 0; p < P; ++p) {
      red[t] = spAcc[(size_t)(gi * P + p) * DDIM + t] * (1.0f / NNODES) *
               att[(size_t)gi * DDIM + t];
      __syncthreads();
      for (int st = 64; st > 0; st >>= 1) {
        if (t < st) red[t] += red[t + st];
        __syncthreads();
      }
      if (t == 0) sc[p] = red[0];
      __syncthreads();
    }
    if (t == 0) {
      float mx = sc[0];
      for (int p = 1; p < P; ++p) mx = fmaxf(mx, sc[p]);
      float sum = 0.0f, ex[8];
      for (int p = 0; p < P; ++p) { ex[p] = __expf(sc[p] - mx); sum += ex[p]; }
      for (int p = 0; p < P; ++p) betaOut[gi * P + p] = ex[p] / sum;
    }
    __syncthreads();
  }
}

// Ob[grp] = beta[grp*2]*Hb[grp*2] + beta[grp*2+1]*Hb[grp*2+1]
__global__ void k_combine2(const float* __restrict__ beta, const float* __restrict__ Hb,
                           float* __restrict__ Ob, long total) {
  const long i = (long)blockIdx.x * blockDim.x + threadIdx.x;
  if (i >= total) return;
  const long nd = (long)NNODES * DDIM;
  const long grp = i / nd;
  const long r = i - grp * nd;
  const float b0 = beta[grp * 2 + 0], b1 = beta[grp * 2 + 1];
  Ob[i] = b0 * Hb[(grp * 2 + 0) * nd + r] + b1 * Hb[(grp * 2 + 1) * nd + r];
}

// ---------------------------------------------------------------------------
extern "C" void kernel_launch(void* const* d_in, const int* in_sizes, int n_in,
                              void* d_out, int out_size, void* d_ws, size_t ws_size,
                              hipStream_t stream) {
  const float* x     = (const float*)d_in[0];
  const int*   src   = (const int*)d_in[1];
  const int*   dst   = (const int*)d_in[2];
  const float* Wsrc  = (const float*)d_in[3];
  const float* bsrc  = (const float*)d_in[4];
  const float* Wdst  = (const float*)d_in[5];
  const float* bdst  = (const float*)d_in[6];
  const float* attn  = (const float*)d_in[7];
  const float* interW = (const float*)d_in[8];
  const float* interB = (const float*)d_in[9];
  const float* interA = (const float*)d_in[10];
  const float* crossW = (const float*)d_in[11];
  const float* crossB = (const float*)d_in[12];
  const float* crossA = (const float*)d_in[13];
  float* outp = (float*)d_out;

  const size_t ND = (size_t)NNODES * DDIM;
  float*    ws        = (float*)d_ws;
  float*    el        = ws;                                   // [G,N,D]
  float*    er        = el + (size_t)NGRAPH * ND;             // [G,N,D], reused as h
  float*    h         = er;
  float*    lemb      = el;                                   // aliases el (free after E4)
  float*    elog      = er + (size_t)NGRAPH * ND;             // [G,E]
  unsigned* emaxU     = (unsigned*)(elog + (size_t)NGRAPH * NEDGES);  // [G,N]
  float*    denom     = (float*)(emaxU + (size_t)NGRAPH * NNODES);    // [G,N]
  float*    spInter   = denom + (size_t)NGRAPH * NNODES;      // [G,D]
  float*    spCross   = spInter + NGRAPH * DDIM;              // [NL,D]
  float*    betaInter = spCross + NLAYER * DDIM;              // [G]
  float*    betaCross = betaInter + NGRAPH;                   // [NL]

  // zero emax/denom/sp/beta region (contiguous)
  {
    long n = 2L * NGRAPH * NNODES + NGRAPH * DDIM + NLAYER * DDIM + NGRAPH + NLAYER;
    k_zero<<<(unsigned)((n + 255) / 256), 256, 0, stream>>>((float*)emaxU, n);
  }
  // el/er projections (WMMA)
  {
    dim3 grid(391, 2 * NGRAPH);
    k_gemm_el_er<<<grid, 256, 0, stream>>>(x, Wsrc, bsrc, Wdst, bdst, el, er);
  }
  // edge logits + segment max
  {
    long th = (long)NGRAPH * NEDGES * 32;
    k_edge_logits<<<(unsigned)((th + 255) / 256), 256, 0, stream>>>(
        el, er, src, dst, attn, elog, emaxU);
  }
  // exp + denom
  {
    long n = (long)NGRAPH * NEDGES;
    k_edge_exp<<<(unsigned)((n + 255) / 256), 256, 0, stream>>>(elog, dst, emaxU, denom);
  }
  // zero h (aliases er, which is no longer needed)
  {
    long n = (long)NGRAPH * ND;
    k_zero<<<(unsigned)((n + 255) / 256), 256, 0, stream>>>(h, n);
  }
  // weighted aggregation
  {
    long th = (long)NGRAPH * NEDGES * 32;
    k_edge_aggregate<<<(unsigned)((th + 255) / 256), 256, 0, stream>>>(
        el, elog, src, dst, denom, h);
  }
  // ELU
  {
    long n = (long)NGRAPH * ND;
    k_elu<<<(unsigned)((n + 255) / 256), 256, 0, stream>>>(h, n);
  }
  // inter-metapath attention scores (WMMA + tanh + mean-reduce)
  {
    dim3 grid(391, NGRAPH);
    k_gemm_tanh_reduce<<<grid, 256, 0, stream>>>(h, interW, interB, spInter, NREL);
  }
  k_beta<<<1, 128, 0, stream>>>(spInter, interA, betaInter, NREL, NLAYER);
  {
    long n = (long)NLAYER * ND;
    k_combine2<<<(unsigned)((n + 255) / 256), 256, 0, stream>>>(betaInter, h, lemb, n);
  }
  // cross-layer attention
  {
    dim3 grid(391, NLAYER);
    k_gemm_tanh_reduce<<<grid, 256, 0, stream>>>(lemb, crossW, crossB, spCross, 16);
  }
  k_beta<<<1, 128, 0, stream>>>(spCross, crossA, betaCross, NLAYER, 1);
  {
    long n = (long)ND;
    k_combine2<<<(unsigned)((n + 255) / 256), 256, 0, stream>>>(betaCross, lemb, outp, n);
  }
}